// CheckinEncoder_12481174962620
// MI455X (gfx1250) — compile-verified
//
#include <hip/hip_runtime.h>
#include <hip/hip_bf16.h>
#include <stdint.h>

// ---------------------------------------------------------------------------
// 2-layer GCN (GCNConv -> PReLU -> GCNConv) for MI455X (gfx1250, wave32).
// GEMMs run on the fp32 WMMA pipe (V_WMMA_F32_16X16X4_F32) to keep the
// reference's fp32 precision. Edge aggregation = float4 gather (L2-resident,
// 192MB L2 >> 102MB feature matrix) + global_atomic_add_f32 scatter.
// ---------------------------------------------------------------------------

typedef float v2f __attribute__((ext_vector_type(2)));
typedef float v8f __attribute__((ext_vector_type(8)));

#define N_NODES 50000
#define N_EDGES 300000
#define IN_CH   256
#define HID     512

// ---------------- degree / norm prep ----------------

__global__ void k_fill_deg(float* __restrict__ deg, int n) {
    int i = blockIdx.x * blockDim.x + threadIdx.x;
    if (i < n) deg[i] = 1.0f;  // self-loop weight
}

__global__ void k_deg_accum(const long long* __restrict__ dst,
                            const float* __restrict__ ew,
                            float* __restrict__ deg, int e) {
    int i = blockIdx.x * blockDim.x + threadIdx.x;
    if (i < e) atomicAdd(deg + (int)dst[i], ew[i]);
}

__global__ void k_dis(const float* __restrict__ deg, float* __restrict__ dis, int n) {
    int i = blockIdx.x * blockDim.x + threadIdx.x;
    if (i < n) {
        float d = deg[i];
        dis[i] = (d > 0.0f) ? rsqrtf(d) : 0.0f;
    }
}

__global__ void k_norm(const long long* __restrict__ src,
                       const long long* __restrict__ dst,
                       const float* __restrict__ ew,
                       const float* __restrict__ dis,
                       float* __restrict__ norm, int e) {
    int i = blockIdx.x * blockDim.x + threadIdx.x;
    if (i < e) norm[i] = dis[(int)src[i]] * ew[i] * dis[(int)dst[i]];
}

// ---------------- fp32 WMMA GEMM:  Y[M,N] = X[M,K] @ W[N,K]^T ----------------
// One wave computes one 16x16 tile; 8 waves/block -> 16x128 strip.
// A-frag (16x4 f32): lane L holds X[m0+(L&15)][k0+kh], X[..][k0+kh+1], kh=(L>>4)*2
// B-frag (4x16 f32): lane L holds W[n0+(L&15)][k0+kh], W[..][k0+kh+1]
// D (16x16 f32):     VGPR v -> row m0 + v + (L>>4)*8, col n0 + (L&15)

__global__ __launch_bounds__(256) void k_gemm_wmma_f32(
    const float* __restrict__ X,   // [M,K]
    const float* __restrict__ W,   // [N,K]
    float* __restrict__ Y,         // [M,N]
    int M, int N, int K)
{
    const int lane = threadIdx.x & 31;
    const int wave = threadIdx.x >> 5;
    const int m0 = blockIdx.x * 16;
    const int n0 = (blockIdx.y * 8 + wave) * 16;

    const int l15 = lane & 15;
    const int kh  = (lane >> 4) * 2;

    const float* ap = X + (size_t)(m0 + l15) * K + kh;
    const float* bp = W + (size_t)(n0 + l15) * K + kh;

    v8f acc = {0.f, 0.f, 0.f, 0.f, 0.f, 0.f, 0.f, 0.f};

#pragma unroll 4
    for (int k0 = 0; k0 < K; k0 += 4) {
        v2f a = *(const v2f*)(ap + k0);
        v2f b = *(const v2f*)(bp + k0);
        // 8 args: (neg_a, A, neg_b, B, c_mod, C, reuse_a, reuse_b)
        acc = __builtin_amdgcn_wmma_f32_16x16x4_f32(
            false, a, false, b, (short)0, acc, false, false);
    }

    const int rbase = m0 + ((lane >> 4) << 3);
    float* yp = Y + (size_t)rbase * N + n0 + l15;
#pragma unroll
    for (int v = 0; v < 8; ++v)
        yp[(size_t)v * N] = acc[v];
}

// ---------------- aggregation ----------------

// agg[n][f] = dis[n]^2 * feat[n][f] + bias[f]   (self-loop + post-agg bias fused)
__global__ void k_init_agg(const float* __restrict__ feat,
                           const float* __restrict__ dis,
                           const float* __restrict__ bias,
                           float* __restrict__ agg, int total4) {
    int t = blockIdx.x * blockDim.x + threadIdx.x;
    if (t >= total4) return;
    int n = t >> 7;              // 128 float4 chunks per node (HID=512)
    int c = (t & 127) << 2;
    float s = dis[n]; s = s * s;
    const float4 v = *(const float4*)(feat + (size_t)n * HID + c);
    const float4 b = *(const float4*)(bias + c);
    float4 o;
    o.x = s * v.x + b.x;
    o.y = s * v.y + b.y;
    o.z = s * v.z + b.z;
    o.w = s * v.w + b.w;
    *(float4*)(agg + (size_t)n * HID + c) = o;
}

// agg[dst] += norm[e] * feat[src]    (one thread = one edge x 4 features)
__global__ void k_scatter(const float* __restrict__ feat,
                          const float* __restrict__ norm,
                          const long long* __restrict__ src,
                          const long long* __restrict__ dst,
                          float* __restrict__ agg, int total4) {
    int t = blockIdx.x * blockDim.x + threadIdx.x;
    if (t >= total4) return;
    int e = t >> 7;
    int c = (t & 127) << 2;
    float nrm = norm[e];
    int s = (int)src[e];
    int d = (int)dst[e];
    const float4 v = *(const float4*)(feat + (size_t)s * HID + c);
    float* o = agg + (size_t)d * HID + c;
    atomicAdd(o + 0, nrm * v.x);
    atomicAdd(o + 1, nrm * v.y);
    atomicAdd(o + 2, nrm * v.z);
    atomicAdd(o + 3, nrm * v.w);
}

__global__ void k_prelu(float* __restrict__ h, const float* __restrict__ pa, int total4) {
    int t = blockIdx.x * blockDim.x + threadIdx.x;
    if (t >= total4) return;
    float a = *pa;
    float4 v = *(float4*)(h + (size_t)t * 4);
    v.x = (v.x >= 0.f) ? v.x : a * v.x;
    v.y = (v.y >= 0.f) ? v.y : a * v.y;
    v.z = (v.z >= 0.f) ? v.z : a * v.z;
    v.w = (v.w >= 0.f) ? v.w : a * v.w;
    *(float4*)(h + (size_t)t * 4) = v;
}

// ---------------------------------------------------------------------------

extern "C" void kernel_launch(void* const* d_in, const int* in_sizes, int n_in,
                              void* d_out, int out_size, void* d_ws, size_t ws_size,
                              hipStream_t stream) {
    const float*     x   = (const float*)d_in[0];
    const long long* ei  = (const long long*)d_in[1];   // int64 [2, E]
    const float*     ew  = (const float*)d_in[2];
    const float*     W1  = (const float*)d_in[3];       // [HID, IN_CH]
    const float*     b1  = (const float*)d_in[4];
    const float*     W2  = (const float*)d_in[5];       // [HID, HID]
    const float*     b2  = (const float*)d_in[6];
    const float*     pa  = (const float*)d_in[7];       // scalar prelu_a

    const int E = in_sizes[2];              // 300000
    const int N = in_sizes[0] / IN_CH;      // 50000

    const long long* src = ei;
    const long long* dst = ei + E;

    // workspace carve-up (256B aligned)
    char* ws = (char*)d_ws;
    size_t off = 0;
    auto carve = [&](size_t bytes) -> char* {
        char* p = ws + off;
        off += (bytes + 255) & ~(size_t)255;
        return p;
    };
    float* xw   = (float*)carve((size_t)N * HID * sizeof(float));  // xw then hw
    float* agg  = (float*)carve((size_t)N * HID * sizeof(float));  // agg1 / h
    float* deg  = (float*)carve((size_t)N * sizeof(float));
    float* dis  = (float*)carve((size_t)N * sizeof(float));
    float* norm = (float*)carve((size_t)E * sizeof(float));
    (void)ws_size;

    const int TB = 256;
    const int nodeBlocks  = (N + TB - 1) / TB;
    const int edgeBlocks  = (E + TB - 1) / TB;
    const int node4       = N * (HID / 4);
    const int edge4       = E * (HID / 4);
    const int node4Blocks = (node4 + TB - 1) / TB;
    const int edge4Blocks = (edge4 + TB - 1) / TB;

    // 1) symmetric-norm coefficients
    k_fill_deg<<<nodeBlocks, TB, 0, stream>>>(deg, N);
    k_deg_accum<<<edgeBlocks, TB, 0, stream>>>(dst, ew, deg, E);
    k_dis<<<nodeBlocks, TB, 0, stream>>>(deg, dis, N);
    k_norm<<<edgeBlocks, TB, 0, stream>>>(src, dst, ew, dis, norm, E);

    // 2) layer 1: xw = x @ W1^T   (M=50000, N=512, K=256)
    {
        dim3 grid(N / 16, HID / (16 * 8));
        k_gemm_wmma_f32<<<grid, TB, 0, stream>>>(x, W1, xw, N, HID, IN_CH);
    }
    k_init_agg<<<node4Blocks, TB, 0, stream>>>(xw, dis, b1, agg, node4);
    k_scatter<<<edge4Blocks, TB, 0, stream>>>(xw, norm, src, dst, agg, edge4);
    k_prelu<<<node4Blocks, TB, 0, stream>>>(agg, pa, node4);

    // 3) layer 2: hw = h @ W2^T   (M=50000, N=512, K=512), hw reuses xw buffer
    {
        dim3 grid(N / 16, HID / (16 * 8));
        k_gemm_wmma_f32<<<grid, TB, 0, stream>>>(agg, W2, xw, N, HID, HID);
    }
    float* out = (float*)d_out;
    k_init_agg<<<node4Blocks, TB, 0, stream>>>(xw, dis, b2, out, node4);
    k_scatter<<<edge4Blocks, TB, 0, stream>>>(xw, norm, src, dst, out, edge4);
}